// batch_get_music_unchunk_1322849927770
// MI455X (gfx1250) — compile-verified
//
#include <hip/hip_runtime.h>

// -----------------------------------------------------------------------------
// Overlap-add reconstruction (gather form), MI455X / gfx1250.
//
// x[B=16, FV=2000, FRAME=2048] f32, HOP=512, PAD0=PAD1=768.
// out[b, j] = ( sum_{k=0..3, 0<=m-k<FV} x[b, m-k, r + 512k] ) / cnt,
//   p = j + 768, m = p>>9, r = p&511.
//
// Memory-bound streaming: 262 MB read + 65.5 MB write, every byte touched
// once -> ~14 us at 23.3 TB/s. Optimal path = coalesced b128 NT loads/stores
// with maximal load batching; no matrix op exists, so no WMMA.
//
// Interior chunks (m in [3,1999], 1997 of 2001): cnt==4 and the 4 taps are
//   xb + m*2048 + r - 1536k  ->  ONE base address + constant IOFFSETs,
// so the 4 global_load_b128 issue back-to-back with a single wait.
// Edge chunks (m in {1,2,2000,2001}): clamped+masked gather, IEEE divide.
// -----------------------------------------------------------------------------

#define FRAME   2048
#define HOP     512
#define PAD0    768
#define NFV     2000
#define OUTLEN  (NFV * HOP)              // 1,024,000 samples per batch row

// Interior j-range: chunks m = 3 .. 1999  ->  j in [768, 1023232)
#define JINT_LO   768
#define NCHUNKS_I 1997                   // 1997 chunks * 512 samples
#define JEDGE_HI  1023232                // start of trailing edge region

typedef float v4f __attribute__((ext_vector_type(4)));

// ---- Interior: one 128-thread block per hop chunk; branch-free, cnt==4 ----
__global__ __launch_bounds__(128) void ola_interior_kernel(
    const float* __restrict__ x,         // [B, NFV, FRAME]
    float* __restrict__ out)             // [B, OUTLEN]
{
    const int m = blockIdx.x + 3;        // hop chunk id (scalar, uniform)
    const int b = blockIdx.y;
    const int r = threadIdx.x << 2;      // 0..508, 16B aligned within chunk

    const float* __restrict__ xb = x + (size_t)b * (size_t)NFV * FRAME;
    // Base for tap k=0; taps k=1..3 are at constant negative element offsets.
    const float* __restrict__ base = xb + (size_t)m * FRAME + (size_t)r;

    // 4 independent b128 NT loads off one address (IOFFSET immediates):
    const v4f v0 = __builtin_nontemporal_load((const v4f*)(base));
    const v4f v1 = __builtin_nontemporal_load((const v4f*)(base - 1536));
    const v4f v2 = __builtin_nontemporal_load((const v4f*)(base - 3072));
    const v4f v3 = __builtin_nontemporal_load((const v4f*)(base - 4608));

    v4f o = (v0 + v1) + (v2 + v3);
    o *= 0.25f;                          // exact: identical to IEEE divide by 4

    const int j = (m << 9) + r - PAD0;   // in [768, 1023232)
    __builtin_nontemporal_store(o, (v4f*)(out + (size_t)b * OUTLEN + (size_t)j));
}

// ---- Edges: 2*768 samples per batch (chunks m = 1,2,2000,2001) ----
__global__ __launch_bounds__(384) void ola_edge_kernel(
    const float* __restrict__ x,
    float* __restrict__ out)
{
    const int b = blockIdx.x;
    const int e = threadIdx.x;           // 0..383 float4s
    const int j = (e < 192) ? (e << 2) : (JEDGE_HI + ((e - 192) << 2));
    const int p = j + PAD0;
    const int m = p >> 9;                // 1,2 or 2000,2001
    const int r = p & (HOP - 1);

    const float* __restrict__ xb = x + (size_t)b * (size_t)NFV * FRAME;

    v4f acc = (v4f){0.0f, 0.0f, 0.0f, 0.0f};
    int cnt = 0;
#pragma unroll
    for (int k = 0; k < 4; ++k) {
        const int f  = m - k;
        // Clamp so the load address is always valid -> loads unconditional
        // (batched); mask out-of-range taps with a 0/1 weight instead.
        const int fc = (f < 0) ? 0 : ((f > NFV - 1) ? (NFV - 1) : f);
        const v4f v = __builtin_nontemporal_load(
            (const v4f*)(xb + (size_t)fc * FRAME + (size_t)(r + (k << 9))));
        const float w = (f == fc) ? 1.0f : 0.0f;
        acc += v * w;
        cnt += (f == fc) ? 1 : 0;
    }

    const float c = (float)cnt;          // 2 or 3 here; never 0
    v4f o;
    o.x = acc.x / c; o.y = acc.y / c; o.z = acc.z / c; o.w = acc.w / c;
    __builtin_nontemporal_store(o, (v4f*)(out + (size_t)b * OUTLEN + (size_t)j));
}

extern "C" void kernel_launch(void* const* d_in, const int* in_sizes, int n_in,
                              void* d_out, int out_size, void* d_ws, size_t ws_size,
                              hipStream_t stream)
{
    (void)n_in; (void)out_size; (void)d_ws; (void)ws_size;
    const float* x = (const float*)d_in[0];
    float* out = (float*)d_out;

    const int B = in_sizes[0] / (NFV * FRAME);   // 16 for reference shapes

    // Interior: 1997 chunks x B batches, 128 threads (one chunk) per block.
    dim3 gi(NCHUNKS_I, B, 1);
    ola_interior_kernel<<<gi, 128, 0, stream>>>(x, out);

    // Edges: one 384-thread block per batch covers the 1536 boundary samples.
    ola_edge_kernel<<<dim3(B), 384, 0, stream>>>(x, out);
}